// PlaceCellNetwork_17927193494329
// MI455X (gfx1250) — compile-verified
//
#include <hip/hip_runtime.h>
#include <math.h>

#define C_ 4
#define IN_ 64
#define OUT_ 128
#define B_ 2048
#define MAXITER_ 100
#define LBD1_ 0.005f
#define LBD2_ 0.005f

typedef float v2f __attribute__((ext_vector_type(2)));
typedef float v8f __attribute__((ext_vector_type(8)));

// fp32 WMMA: D(16x16) = A(16x4) * B(4x16) + C   (one tile per wave32)
__device__ __forceinline__ v8f wmma4(v2f a, v2f b, v8f c) {
  return __builtin_amdgcn_wmma_f32_16x16x4_f32(
      /*neg_a=*/false, a, /*neg_b=*/false, b,
      /*c_mod=*/(short)0, c, /*reuse_a=*/false, /*reuse_b=*/false);
}

// ---------------------------------------------------------------------------
// init: Y0 = 0, Moff = M - I, diagM = diag(M), copy errTrack, clear accum/done
// ---------------------------------------------------------------------------
__global__ void init_kernel(const float* __restrict__ M,
                            const float* __restrict__ eT_in,
                            float* __restrict__ Moff,
                            float* __restrict__ diagM,
                            float* __restrict__ eT,
                            float* __restrict__ accum,
                            int* __restrict__ done,
                            float* __restrict__ Y0) {
  int gtid = blockIdx.x * blockDim.x + threadIdx.x;
  int stride = gridDim.x * blockDim.x;
  for (int i = gtid; i < C_ * B_ * OUT_; i += stride) Y0[i] = 0.f;
  for (int i = gtid; i < C_ * OUT_ * OUT_; i += stride) {
    int rem = i % (OUT_ * OUT_);
    int p = rem / OUT_, o = rem % OUT_;
    float m = M[i];
    Moff[i] = m - ((p == o) ? 1.f : 0.f);
    if (p == o) diagM[(i / (OUT_ * OUT_)) * OUT_ + p] = m;
  }
  if (gtid < C_ * 5) eT[gtid] = eT_in[gtid];
  if (gtid < 2 * C_) accum[gtid] = 0.f;
  if (gtid == 0) *done = 0;
}

// ---------------------------------------------------------------------------
// Wx[c,b,o] = sum_i X[b,i] * W[c,o,i]   (A = X tile, B[k][n] = W[c][n][k])
// one 16x16 output tile per wave; K = 64 -> 16 x v_wmma_f32_16x16x4_f32
// ---------------------------------------------------------------------------
__global__ void __launch_bounds__(256) wx_kernel(const float* __restrict__ X,
                                                 const float* __restrict__ W,
                                                 float* __restrict__ Wx) {
  int wid  = (blockIdx.x * blockDim.x + threadIdx.x) >> 5;
  int lane = threadIdx.x & 31;
  int c  = wid >> 10;            // tiles: 128 (b) * 8 (o) per channel
  int rm = wid & 1023;
  int tb = rm >> 3;
  int to = rm & 7;
  int nrow = lane & 15;          // A row / B col within tile
  int kh   = (lane >> 4) << 1;   // K sub-offset per half-wave
  const float* Arow = X + (size_t)(tb * 16 + nrow) * IN_;
  const float* Brow = W + (size_t)(c * OUT_ + to * 16 + nrow) * IN_;
  v8f acc = {};
#pragma unroll
  for (int k0 = 0; k0 < IN_; k0 += 4) {
    v2f a = *(const v2f*)(Arow + k0 + kh);
    v2f b = *(const v2f*)(Brow + k0 + kh);
    acc = wmma4(a, b, acc);
  }
  int h = lane >> 4;
  int o = to * 16 + nrow;
#pragma unroll
  for (int v = 0; v < 8; ++v) {
    int r = tb * 16 + v + 8 * h;
    Wx[(size_t)(c * B_ + r) * OUT_ + o] = acc[v];
  }
}

// ---------------------------------------------------------------------------
// one dynamics step, fused:
//   MY = Yold * Moff^T (WMMA, K=128 -> 32 wmma/tile)
//   uy = Yold + dt*(-Yold + Wx - b - MY)
//   Ynew = max((uy - l1)/(l2 + diagM), 0)
//   accumulate sum(dY^2), sum(Yold^2) per channel (wave shuffle + atomics)
// ---------------------------------------------------------------------------
__global__ void __launch_bounds__(256) iter_kernel(
    const float* __restrict__ Yold, float* __restrict__ Ynew,
    float* __restrict__ out, const float* __restrict__ Wx,
    const float* __restrict__ Moff, const float* __restrict__ diagM,
    const float* __restrict__ bvec, float* __restrict__ accum,
    const int* __restrict__ done, float dt) {
  if (*done) return;  // wave-uniform: EXEC stays all-ones past this point
  int wid  = (blockIdx.x * blockDim.x + threadIdx.x) >> 5;
  int lane = threadIdx.x & 31;
  int c  = wid >> 10;
  int rm = wid & 1023;
  int tb = rm >> 3;
  int to = rm & 7;
  int nrow = lane & 15;
  int kh   = (lane >> 4) << 1;
  const float* Arow = Yold + (size_t)(c * B_ + tb * 16 + nrow) * OUT_;
  const float* Brow = Moff + (size_t)(c * OUT_ + to * 16 + nrow) * OUT_;
  v8f acc = {};
#pragma unroll 8
  for (int k0 = 0; k0 < OUT_; k0 += 4) {
    v2f a = *(const v2f*)(Arow + k0 + kh);
    v2f b = *(const v2f*)(Brow + k0 + kh);
    acc = wmma4(a, b, acc);
  }
  int h = lane >> 4;
  int o = to * 16 + nrow;
  float bv  = bvec[c * OUT_ + o];           // sqrt(ALPHA) == 1
  float inv = 1.f / (LBD2_ + diagM[c * OUT_ + o]);
  float s_d2 = 0.f, s_y2 = 0.f;
#pragma unroll
  for (int v = 0; v < 8; ++v) {
    size_t idx = (size_t)(c * B_ + tb * 16 + v + 8 * h) * OUT_ + o;
    float yo = Yold[idx];
    float wx = Wx[idx];
    float uy = yo + dt * (wx - yo - bv - acc[v]);
    float yn = (uy - LBD1_) * inv;
    yn = yn > 0.f ? yn : 0.f;
    Ynew[idx] = yn;
    out[idx]  = yn;                          // d_out always holds latest Y
    float d = yn - yo;
    s_d2 += d * d;
    s_y2 += yo * yo;
  }
#pragma unroll
  for (int off = 16; off > 0; off >>= 1) {
    s_d2 += __shfl_xor(s_d2, off, 32);
    s_y2 += __shfl_xor(s_y2, off, 32);
  }
  if (lane == 0) {
    atomicAdd(&accum[2 * c + 0], s_d2);
    atomicAdd(&accum[2 * c + 1], s_y2);
  }
}

// ---------------------------------------------------------------------------
// convergence bookkeeping (matches jax while_loop body tail)
// ---------------------------------------------------------------------------
__global__ void finalize_kernel(float* __restrict__ accum,
                                float* __restrict__ eT,
                                int* __restrict__ done, float dt) {
  if (threadIdx.x != 0 || blockIdx.x != 0) return;
  if (*done) return;
  bool allErr = true, allCum = true;
  for (int c = 0; c < C_; ++c) {
    float err = sqrtf(accum[2 * c]) / (sqrtf(accum[2 * c + 1]) + 1e-10f) / dt;
    float cum = fabsf(eT[c * 5 + 3] - err);   // |eT_new[4] - eT_new[0]|
    eT[c * 5 + 4] = eT[c * 5 + 3];
    eT[c * 5 + 3] = eT[c * 5 + 2];
    eT[c * 5 + 2] = eT[c * 5 + 1];
    eT[c * 5 + 1] = eT[c * 5 + 0];
    eT[c * 5 + 0] = err;
    allErr = allErr && (err < 1e-4f);
    allCum = allCum && (cum < 1e-8f);
    accum[2 * c] = 0.f;
    accum[2 * c + 1] = 0.f;
  }
  if (allErr && allCum) *done = 1;
}

extern "C" void kernel_launch(void* const* d_in, const int* in_sizes, int n_in,
                              void* d_out, int out_size, void* d_ws, size_t ws_size,
                              hipStream_t stream) {
  const float* X     = (const float*)d_in[0];   // [B, IN]
  const float* W     = (const float*)d_in[1];   // [C, OUT, IN]
  const float* M     = (const float*)d_in[2];   // [C, OUT, OUT]
  const float* bvec  = (const float*)d_in[3];   // [C, OUT]
  const float* eT_in = (const float*)d_in[4];   // [C, 5, 1]
  float* out = (float*)d_out;                   // [C, B, OUT]
  (void)in_sizes; (void)n_in; (void)out_size; (void)ws_size;

  float* wsf = (float*)d_ws;
  size_t off = 0;
  float* Moff  = wsf + off; off += (size_t)C_ * OUT_ * OUT_;   // 256 KB
  float* diagM = wsf + off; off += (size_t)C_ * OUT_;
  float* eT    = wsf + off; off += 64;                         // C*5 padded
  float* accum = wsf + off; off += 64;                         // 2*C padded
  int*   done  = (int*)(wsf + off); off += 64;
  float* Wx    = wsf + off; off += (size_t)C_ * B_ * OUT_;     // 4 MB
  float* Ybuf0 = wsf + off; off += (size_t)C_ * B_ * OUT_;     // 4 MB
  float* Ybuf1 = wsf + off; off += (size_t)C_ * B_ * OUT_;     // 4 MB

  init_kernel<<<1024, 256, 0, stream>>>(M, eT_in, Moff, diagM, eT, accum, done, Ybuf0);

  // 4096 tiles (16x16) -> 4096 waves -> 512 blocks of 8 waves
  wx_kernel<<<512, 256, 0, stream>>>(X, W, Wx);

  for (int it = 0; it < MAXITER_; ++it) {
    float dt = fmaxf(0.05f / (1.f + (float)it / 10.f), 0.01f);
    const float* Yold = (it & 1) ? Ybuf1 : Ybuf0;
    float* Ynew       = (it & 1) ? Ybuf0 : Ybuf1;
    iter_kernel<<<512, 256, 0, stream>>>(Yold, Ynew, out, Wx, Moff, diagM,
                                         bvec, accum, done, dt);
    finalize_kernel<<<1, 32, 0, stream>>>(accum, eT, done, dt);
  }
}